// clustering_dynamic_learning_common_center_5_45286135169479
// MI455X (gfx1250) — compile-verified
//
#include <hip/hip_runtime.h>
#include <hip/hip_bf16.h>

// Problem constants
#define B_ 32
#define N_ 2000
#define K_ 32
#define C_ 8
#define F_ 64
#define T_ 12
#define S_ 12

typedef __attribute__((ext_vector_type(2))) float v2f;
typedef __attribute__((ext_vector_type(8))) float v8f;

// Workspace layout (float offsets)
#define OFF_SCALE 0          // 2000
#define OFF_SHIFT 2000       // 2000
#define OFF_ADJ   4000       // 32*64
#define OFF_N2C   6048       // 32*8
#define OFF_SA    6304       // 32*2000*8
#define OFF_LSM   518304     // 32*2000*8
#define OFF_CR    1030304    // 2000*32*8
#define OFF_H     1542304    // 32*2000*96
#define OFF_PCENT 7686304    // 32
#define OFF_PKL   7686336    // 32
#define OFF_PA1   7686368    // 250
#define OFF_PS2   7686618    // 4000

// ---------------- K0: batchnorm stats per n over (B,F) ----------------
__global__ void k_bn_stats(const float* __restrict__ fused,
                           const float* __restrict__ gamma,
                           const float* __restrict__ beta,
                           float* __restrict__ scale,
                           float* __restrict__ shift) {
    __shared__ float s1[256], s2[256];
    int n = blockIdx.x, t = threadIdx.x;
    float a = 0.f, q = 0.f;
    for (int i = t; i < B_ * F_; i += 256) {
        int bb = i >> 6, f = i & 63;
        float v = fused[((size_t)bb * N_ + n) * F_ + f];
        a += v; q += v * v;
    }
    s1[t] = a; s2[t] = q; __syncthreads();
    for (int o = 128; o > 0; o >>= 1) {
        if (t < o) { s1[t] += s1[t + o]; s2[t] += s2[t + o]; }
        __syncthreads();
    }
    if (t == 0) {
        float mean = s1[0] / (float)(B_ * F_);
        float var  = s2[0] / (float)(B_ * F_) - mean * mean;
        float sc = gamma[n] * rsqrtf(var + 1e-5f);
        scale[n] = sc;
        shift[n] = beta[n] - mean * sc;
    }
}

// ---------------- K1: adj[b][f] = mean_n ff[b,n,f] ----------------
__global__ void k_adjv(const float* __restrict__ fused,
                       const float* __restrict__ scale,
                       const float* __restrict__ shift,
                       float* __restrict__ adjv) {
    int b = blockIdx.x, f = threadIdx.x;
    float sA = 0.f, sS = 0.f;
    for (int n = 0; n < N_; ++n) {
        sA += fused[((size_t)b * N_ + n) * F_ + f] * scale[n];
        sS += shift[n];
    }
    adjv[b * F_ + f] = (sA + sS) / (float)N_;
}

// ---------------- K2: centroid MLP + centroid losses + n2c ----------------
__global__ void k_centroid(const float* __restrict__ cenG, const float* __restrict__ adjv,
                           const float* __restrict__ W1, const float* __restrict__ b1,
                           const float* __restrict__ W2, const float* __restrict__ b2,
                           const float* __restrict__ Wc, const float* __restrict__ bc,
                           float* __restrict__ n2c, float* __restrict__ pcent,
                           float* __restrict__ pkl) {
    __shared__ float cen[512], cc[512], p1[512], csb[512], cfs[512], lcs[512];
    __shared__ float n2r[8], rmax[8], rlse[8], red[64];
    int b = blockIdx.x, t = threadIdx.x;
    for (int i = t; i < 512; i += 64) cen[i] = cenG[(size_t)b * 512 + i];
    __syncthreads();
    // n2 of (centroid - adj_ff) for K3's cdist(ff, centroids)
    if (t < 8) {
        float s = 0.f;
        for (int f = 0; f < 64; ++f) { float d = cen[t * 64 + f] - adjv[b * 64 + f]; s += d * d; }
        n2c[b * 8 + t] = s;
    }
    // center columns by centroid row-mean (for cdist(centroids, centroids))
    {
        float m = 0.f;
        for (int c = 0; c < 8; ++c) m += cen[c * 64 + t];
        m *= 0.125f;
        for (int c = 0; c < 8; ++c) cc[c * 64 + t] = cen[c * 64 + t] - m;
    }
    __syncthreads();
    if (t < 8) {
        float s = 0.f;
        for (int f = 0; f < 64; ++f) s += cc[t * 64 + f] * cc[t * 64 + f];
        n2r[t] = s;
    }
    __syncthreads();
    // pairwise centroid margin loss
    {
        int i = t >> 3, j = t & 7;
        float dot = 0.f;
        for (int f = 0; f < 64; ++f) dot += cc[i * 64 + f] * cc[j * 64 + f];
        float dist = sqrtf(fmaxf(n2r[i] + n2r[j] - 2.f * dot, 1e-30f));
        float tgt = (i != j) ? 1.f : 0.f;
        float d = fmaxf(tgt - dist, 0.f);
        red[t] = d * d;
    }
    __syncthreads();
    for (int o = 32; o > 0; o >>= 1) { if (t < o) red[t] += red[t + o]; __syncthreads(); }
    if (t == 0) pcent[b] = red[0];
    __syncthreads();
    // p1 = relu(cen @ W1^T + b1)
    for (int c = 0; c < 8; ++c) {
        float s = b1[t];
        for (int f = 0; f < 64; ++f) s += cen[c * 64 + f] * W1[t * 64 + f];
        p1[c * 64 + t] = fmaxf(s, 0.f);
    }
    __syncthreads();
    // cf = relu(p1 @ W2^T + b2) + relu(cen @ Wc^T + bc)
    for (int c = 0; c < 8; ++c) {
        float s = b2[t];
        for (int f = 0; f < 64; ++f) s += p1[c * 64 + f] * W2[t * 64 + f];
        float h1v = fmaxf(s, 0.f);
        float s2v = bc[t];
        for (int f = 0; f < 64; ++f) s2v += cen[c * 64 + f] * Wc[t * 64 + f];
        cfs[c * 64 + t] = h1v + fmaxf(s2v, 0.f);
    }
    __syncthreads();
    if (t < 8) {
        float mx = -1e30f;
        for (int f = 0; f < 64; ++f) mx = fmaxf(mx, cfs[t * 64 + f]);
        float se = 0.f;
        for (int f = 0; f < 64; ++f) se += expf(cfs[t * 64 + f] - mx);
        rmax[t] = mx; rlse[t] = logf(se);
    }
    __syncthreads();
    for (int c = 0; c < 8; ++c) {
        float l = cfs[c * 64 + t] - rmax[c] - rlse[c];
        lcs[c * 64 + t] = l;
        csb[c * 64 + t] = expf(l);
    }
    __syncthreads();
    // dkl[i][j] = sum_f cs[j]*(lcs[j]-lcs[i]); off-diagonal sum
    {
        int i = t >> 3, j = t & 7;
        float s = 0.f;
        if (i != j)
            for (int f = 0; f < 64; ++f) s += csb[j * 64 + f] * (lcs[j * 64 + f] - lcs[i * 64 + f]);
        red[t] = s;
    }
    __syncthreads();
    for (int o = 32; o > 0; o >>= 1) { if (t < o) red[t] += red[t + o]; __syncthreads(); }
    if (t == 0) pkl[b] = red[0];
}

// ---------------- K3: simi -> thresholded softmax (sa) + log-softmax (lsm) ----------------
__global__ void k_simi(const float* __restrict__ fused, const float* __restrict__ cenG,
                       const float* __restrict__ adjv, const float* __restrict__ scale,
                       const float* __restrict__ shift, const float* __restrict__ n2c,
                       float* __restrict__ sa, float* __restrict__ lsm) {
    __shared__ float cencl[512], adjl[64], n2l[8];
    int b = blockIdx.y, t = threadIdx.x;
    for (int i = t; i < 512; i += 128) {
        int f = i & 63;
        cencl[i] = cenG[(size_t)b * 512 + i] - adjv[b * 64 + f];
    }
    if (t < 64) adjl[t] = adjv[b * 64 + t];
    if (t < 8)  n2l[t]  = n2c[b * 8 + t];
    __syncthreads();
    int n = blockIdx.x * 128 + t;
    if (n >= N_) return;
    float sc = scale[n], sh = shift[n];
    const float* xr = fused + ((size_t)b * N_ + n) * 64;
    float n1 = 0.f, dot[8];
#pragma unroll
    for (int c = 0; c < 8; ++c) dot[c] = 0.f;
    for (int f = 0; f < 64; ++f) {
        float xc = xr[f] * sc + sh - adjl[f];
        n1 += xc * xc;
#pragma unroll
        for (int c = 0; c < 8; ++c) dot[c] += xc * cencl[c * 64 + f];
    }
    float si[8], mx = -1e30f;
#pragma unroll
    for (int c = 0; c < 8; ++c) {
        si[c] = sqrtf(fmaxf(n1 + n2l[c] - 2.f * dot[c], 1e-30f));
        mx = fmaxf(mx, si[c]);
    }
    float se = 0.f;
#pragma unroll
    for (int c = 0; c < 8; ++c) { float e = expf(si[c] - mx); si[c] = e; se += e; }
    float inv = 1.f / se;
#pragma unroll
    for (int c = 0; c < 8; ++c) { float p = si[c] * inv; si[c] = (p < 0.125f) ? 0.f : p; }
    float mx2 = -1e30f;
#pragma unroll
    for (int c = 0; c < 8; ++c) mx2 = fmaxf(mx2, si[c]);
    float se2 = 0.f;
#pragma unroll
    for (int c = 0; c < 8; ++c) se2 += expf(si[c] - mx2);
    float l2 = logf(se2);
    size_t o = ((size_t)b * N_ + n) * 8;
#pragma unroll
    for (int c = 0; c < 8; ++c) { sa[o + c] = si[c]; lsm[o + c] = si[c] - mx2 - l2; }
}

// ---------------- Kcr: cr = softmax(cluster_results); A1 partials ----------------
__global__ void k_cr(const float* __restrict__ clr, float* __restrict__ cr,
                     float* __restrict__ pA1) {
    __shared__ float red[256];
    int t = threadIdx.x;
    int row = blockIdx.x * 256 + t;   // 250*256 == 64000 exactly
    float v[8], mx = -1e30f;
#pragma unroll
    for (int c = 0; c < 8; ++c) { v[c] = clr[(size_t)row * 8 + c]; mx = fmaxf(mx, v[c]); }
    float se = 0.f;
#pragma unroll
    for (int c = 0; c < 8; ++c) { float e = expf(v[c] - mx); v[c] = e; se += e; }
    float inv = 1.f / se, a1 = 0.f;
#pragma unroll
    for (int c = 0; c < 8; ++c) {
        float p = v[c] * inv;
        cr[(size_t)row * 8 + c] = p;
        a1 += p * logf(p);
    }
    red[t] = a1; __syncthreads();
    for (int o = 128; o > 0; o >>= 1) { if (t < o) red[t] += red[t + o]; __syncthreads(); }
    if (t == 0) pA1[blockIdx.x] = red[0];
}

// ---------------- K4: H[b,m,c*12+s] = relu(x @ Wt + bias) via FP32 WMMA ----------------
// rows = B*N = 64000, cols = C*S = 96, K = T = 12 (3 chained 16x16x4 f32 WMMAs)
__global__ void k_hgemm(const float* __restrict__ x, const float* __restrict__ gW,
                        const float* __restrict__ gB, float* __restrict__ H) {
    __shared__ float Wt[12 * 96];
    __shared__ float bias[96];
    int t = threadIdx.x;
    for (int i = t; i < C_ * S_ * T_; i += 128) {
        int c = i / (S_ * T_);
        int rem = i - c * (S_ * T_);
        int s = rem / T_;
        int tt = rem - s * T_;
        Wt[tt * 96 + c * 12 + s] = gW[i];   // transpose to (T, C*S)
    }
    if (t < 96) bias[t] = gB[t];            // gcn_b (C,S) flat == c*12+s
    __syncthreads();

    int wave = t >> 5, lane = t & 31;
    int r0 = (blockIdx.x * 4 + wave) * 16;  // 1000 blocks * 4 waves * 16 rows = 64000
    int ml = lane & 15;
    int koff = (lane >> 4) << 1;            // lanes 0-15: K=k0,k0+1 ; lanes 16-31: K=k0+2,k0+3
    int hiRow = (lane >> 4) * 8;            // D layout: VGPR r -> rows r / r+8
    const float* xr = x + (size_t)(r0 + ml) * T_;
    v2f a0 = { xr[0 + koff], xr[1 + koff] };
    v2f a1 = { xr[4 + koff], xr[5 + koff] };
    v2f a2 = { xr[8 + koff], xr[9 + koff] };
#pragma unroll
    for (int j = 0; j < 6; ++j) {
        int col = j * 16 + ml;
        v2f w0 = { Wt[(0 + koff) * 96 + col], Wt[(1 + koff) * 96 + col] };
        v2f w1 = { Wt[(4 + koff) * 96 + col], Wt[(5 + koff) * 96 + col] };
        v2f w2 = { Wt[(8 + koff) * 96 + col], Wt[(9 + koff) * 96 + col] };
        v8f acc = {};
        acc = __builtin_amdgcn_wmma_f32_16x16x4_f32(false, a0, false, w0, (short)0, acc, false, false);
        acc = __builtin_amdgcn_wmma_f32_16x16x4_f32(false, a1, false, w1, (short)0, acc, false, false);
        acc = __builtin_amdgcn_wmma_f32_16x16x4_f32(false, a2, false, w2, (short)0, acc, false, false);
        float bv = bias[col];
#pragma unroll
        for (int r = 0; r < 8; ++r) {
            float h = fmaxf(acc[r] + bv, 0.f);
            H[(size_t)(r0 + r + hiRow) * 96 + col] = h;
        }
    }
}

// ---------------- K5: gather over K neighbors, weighted mean, KL partials ----------------
__global__ void k_gather(const float* __restrict__ H, const float* __restrict__ sa,
                         const float* __restrict__ lsm, const float* __restrict__ cr,
                         const int* __restrict__ adj, float* __restrict__ out,
                         float* __restrict__ pS2) {
    __shared__ int lidx[16 * 32];
    __shared__ float s2red[16];
    int t = threadIdx.x;
    int base = blockIdx.x * 16;             // 4000 blocks * 16 pairs = 64000 (b,n) pairs
    for (int i = t; i < 512; i += 192) {
        int gp = base + (i >> 5);
        lidx[i] = adj[(size_t)gp * 32 + (i & 31)];   // (B,N,K) flat == gp*32+k
    }
    __syncthreads();
    int pl = t / 12, s = t - pl * 12;
    int gp = base + pl;
    int b = gp / N_;
    int n = gp - b * N_;
    float num[8], cnt[8];
#pragma unroll
    for (int c = 0; c < 8; ++c) { num[c] = 0.f; cnt[c] = 0.f; }
    float s2 = 0.f;
    for (int k = 0; k < 32; ++k) {
        int m = lidx[pl * 32 + k];
        size_t gm = (size_t)b * N_ + m;
        if (k + 1 < 32) {
            int mn = lidx[pl * 32 + k + 1];
            __builtin_prefetch(H + ((size_t)b * N_ + mn) * 96, 0, 0);
        }
        const float* hr = H + gm * 96 + s;
        const float* sr = sa + gm * 8;
#pragma unroll
        for (int c = 0; c < 8; ++c) {
            float w = sr[c];
            num[c] += w * hr[c * 12];
            cnt[c] += w;
        }
        if (s == 0) {
            const float* lr = lsm + gm * 8;
            const float* crr = cr + ((size_t)n * 32 + k) * 8;
#pragma unroll
            for (int c = 0; c < 8; ++c) s2 += crr[c] * lr[c];
        }
    }
#pragma unroll
    for (int c = 0; c < 8; ++c) {
        float ct = cnt[c];
        ct = (ct == 0.f) ? 1.f : ct;
        out[((size_t)gp * 8 + c) * 12 + s] = num[c] / ct;
    }
    if (s == 0) s2red[pl] = s2;
    __syncthreads();
    if (t == 0) {
        float tot = 0.f;
        for (int i = 0; i < 16; ++i) tot += s2red[i];
        pS2[blockIdx.x] = tot;
    }
}

// ---------------- K6: deterministic final reductions -> scalar outputs ----------------
__global__ void k_final(const float* __restrict__ pA1, const float* __restrict__ pS2,
                        const float* __restrict__ pc, const float* __restrict__ pk,
                        float* __restrict__ out) {
    if (threadIdx.x != 0 || blockIdx.x != 0) return;
    float A1 = 0.f; for (int i = 0; i < 250; ++i)  A1 += pA1[i];
    float S2 = 0.f; for (int i = 0; i < 4000; ++i) S2 += pS2[i];
    float ce = 0.f, kl = 0.f;
    for (int i = 0; i < 32; ++i) { ce += pc[i]; kl += pk[i]; }
    out[(size_t)B_ * N_ * C_ * S_ + 0] = ((float)B_ * A1 - S2) / (float)(B_ * N_);
    out[(size_t)B_ * N_ * C_ * S_ + 1] = ce / (float)B_;
    out[(size_t)B_ * N_ * C_ * S_ + 2] = -(kl / (float)B_);
}

extern "C" void kernel_launch(void* const* d_in, const int* in_sizes, int n_in,
                              void* d_out, int out_size, void* d_ws, size_t ws_size,
                              hipStream_t stream) {
    const float* fused  = (const float*)d_in[0];
    const float* xin    = (const float*)d_in[1];   // (B,1,N,T) -> (B*N,12)
    const int*   adj    = (const int*)d_in[2];
    const float* cen    = (const float*)d_in[3];
    const float* W1     = (const float*)d_in[4];
    const float* b1     = (const float*)d_in[5];
    const float* W2     = (const float*)d_in[6];
    const float* b2     = (const float*)d_in[7];
    const float* Wc     = (const float*)d_in[8];
    const float* bc     = (const float*)d_in[9];
    const float* gW     = (const float*)d_in[10];
    const float* gB     = (const float*)d_in[11];
    const float* clr    = (const float*)d_in[12];
    const float* gamma  = (const float*)d_in[13];
    const float* beta   = (const float*)d_in[14];
    float* out = (float*)d_out;
    float* ws  = (float*)d_ws;

    float* scale = ws + OFF_SCALE;
    float* shift = ws + OFF_SHIFT;
    float* adjv  = ws + OFF_ADJ;
    float* n2c   = ws + OFF_N2C;
    float* sa    = ws + OFF_SA;
    float* lsm   = ws + OFF_LSM;
    float* cr    = ws + OFF_CR;
    float* H     = ws + OFF_H;
    float* pcent = ws + OFF_PCENT;
    float* pkl   = ws + OFF_PKL;
    float* pA1   = ws + OFF_PA1;
    float* pS2   = ws + OFF_PS2;

    k_bn_stats<<<N_, 256, 0, stream>>>(fused, gamma, beta, scale, shift);
    k_adjv<<<B_, 64, 0, stream>>>(fused, scale, shift, adjv);
    k_centroid<<<B_, 64, 0, stream>>>(cen, adjv, W1, b1, W2, b2, Wc, bc, n2c, pcent, pkl);
    k_simi<<<dim3((N_ + 127) / 128, B_), 128, 0, stream>>>(fused, cen, adjv, scale, shift, n2c, sa, lsm);
    k_cr<<<250, 256, 0, stream>>>(clr, cr, pA1);
    k_hgemm<<<1000, 128, 0, stream>>>(xin, gW, gB, H);
    k_gather<<<4000, 192, 0, stream>>>(H, sa, lsm, cr, adj, out, pS2);
    k_final<<<1, 1, 0, stream>>>(pA1, pS2, pcent, pkl, out);
}